// SinkhornLayer_46213848105379
// MI455X (gfx1250) — compile-verified
//
#include <hip/hip_runtime.h>

// Sinkhorn on 4096 independent 128x128 matrices, reformulated from log-domain
// (logsumexp subtractions) to the exactly-equivalent probability-domain
// iteration (divide by row/col sums), eliminating 2.7e9 transcendental ops.
// Row sums use v_wmma_f32_16x16x4_f32 with an all-ones B matrix
// (layout-proof: D[m][n] = sum_k A[m][k]).

constexpr int kN     = 128;   // matrix dim
constexpr int kW     = 129;   // padded LDS stride (conflict-free rows & cols)
constexpr int kIters = 21;

typedef float v2f __attribute__((ext_vector_type(2)));
typedef float v8f __attribute__((ext_vector_type(8)));

#if __has_builtin(__builtin_amdgcn_wmma_f32_16x16x4_f32)
#define HAVE_WMMA_F32 1
#endif

#if __has_builtin(__builtin_amdgcn_global_load_async_to_lds_b32)
#define HAVE_ASYNC_LDS 1
typedef __attribute__((address_space(1))) int gint_t;   // global (AS1)
typedef __attribute__((address_space(3))) int lint_t;   // LDS (AS3)
#endif

// Single v_rcp_f32 (~1 ulp) instead of the full IEEE divide expansion.
__device__ __forceinline__ float fast_rcp(float x) {
#if __has_builtin(__builtin_amdgcn_rcpf)
  return __builtin_amdgcn_rcpf(x);
#else
  return 1.0f / x;
#endif
}

__global__ __launch_bounds__(kN) void sinkhorn128_kernel(
    const float* __restrict__ in, float* __restrict__ out) {
  __shared__ float M[kN * kW];   // tile, padded
  __shared__ float rrec[kN];     // reciprocal row sums

  const int tid  = threadIdx.x;        // 0..127
  const int lane = tid & 31;           // wave32 lane
  const int wave = tid >> 5;           // 0..3

  const float* gin  = in  + (size_t)blockIdx.x * (kN * kN);
  float*       gout = out + (size_t)blockIdx.x * (kN * kN);

  // ---- Stage 128x128 tile into padded LDS (async DMA path on gfx1250) ----
  for (int k = 0; k < kN; ++k) {
    const int lidx = k * kW + tid;   // row k, col tid (padded)
    const int gidx = k * kN + tid;   // coalesced: 512B per row per k
#if defined(HAVE_ASYNC_LDS)
    __builtin_amdgcn_global_load_async_to_lds_b32(
        (gint_t*)(gin + gidx), (lint_t*)(&M[lidx]), 0, 0);
#else
    M[lidx] = gin[gidx];
#endif
  }
#if defined(HAVE_ASYNC_LDS)
#if __has_builtin(__builtin_amdgcn_s_wait_asynccnt)
  __builtin_amdgcn_s_wait_asynccnt(0);
#else
  asm volatile("s_wait_asynccnt 0" ::: "memory");
#endif
#endif
  __syncthreads();

  // ---- One-time exp with per-row max subtraction (thread t owns row t).
  // exp(x/T - max/T) only rescales the row; the first row-normalization
  // cancels the scaling, so this matches the reference exactly while
  // keeping exp() in range.  Bank = (row + col) mod 64 -> conflict-free.
  {
    const int   base = tid * kW;
    const float invT = 1.0f / 0.04f;
    float m = -3.402823466e38f;
    for (int c = 0; c < kN; ++c) m = fmaxf(m, M[base + c]);
    for (int c = 0; c < kN; ++c) {
      float v = M[base + c];
      M[base + c] = __expf((v - m) * invT);
    }
  }
  __syncthreads();

  v2f onesB;
  onesB.x = 1.0f;
  onesB.y = 1.0f;

  for (int it = 0; it < kIters; ++it) {
    // ---- (1) Row sums on the matrix core: D = A x J (J = ones 4x16).
    // A-layout (16x4 f32): lanes 0-15 -> (M=lane, K=0|1), lanes 16-31 ->
    // (M=lane-16, K=2|3).  Chain K=128 as 32 x (16x16x4) WMMAs per strip.
#if defined(HAVE_WMMA_F32)
    for (int s = wave; s < 8; s += 4) {         // 8 strips of 16 rows
      const int r0    = (s << 4) + (lane & 15);
      const int koff  = (lane >> 4) << 1;       // 0 or 2
      const int abase = r0 * kW + koff;
      v8f acc = {};
#pragma unroll 8
      for (int k4 = 0; k4 < 32; ++k4) {
        v2f a;
        a.x = M[abase + (k4 << 2)];
        a.y = M[abase + (k4 << 2) + 1];
        acc = __builtin_amdgcn_wmma_f32_16x16x4_f32(
            false, a, false, onesB, (short)0, acc, false, false);
      }
      // C/D layout: VGPR v holds row 16s+v (lanes 0-15) / 16s+8+v (16-31).
      if ((lane & 15) == 0) {
        const int rbase = (s << 4) + ((lane >> 4) << 3);
#pragma unroll
        for (int v = 0; v < 8; ++v)
          rrec[rbase + v] = fast_rcp(fmaxf(acc[v], 1e-38f));
      }
    }
#else
    {  // scalar fallback: thread t reduces row t
      float sr = 0.0f;
      for (int c = 0; c < kN; ++c) sr += M[tid * kW + c];
      rrec[tid] = fast_rcp(fmaxf(sr, 1e-38f));
    }
#endif
    __syncthreads();

    // ---- (2) Scale rows by 1/rowsum, fused with column-sum accumulation.
    // Thread t owns column t: LDS bank (i + t) mod 64 -> conflict-free;
    // rrec[i] is a same-address broadcast read.
    float cs = 0.0f;
    for (int i = 0; i < kN; ++i) {
      float v = M[i * kW + tid] * rrec[i];
      M[i * kW + tid] = v;
      cs += v;
    }
    const float invc = fast_rcp(fmaxf(cs, 1e-38f));

    // ---- (3) Scale column by 1/colsum (no cross-thread dep with (2)).
    for (int i = 0; i < kN; ++i) M[i * kW + tid] *= invc;
    __syncthreads();
  }

  // ---- Output is already exp(final log_alpha); coalesced store.
  for (int k = 0; k < kN; ++k) gout[k * kN + tid] = M[k * kW + tid];
}

extern "C" void kernel_launch(void* const* d_in, const int* in_sizes, int n_in,
                              void* d_out, int out_size, void* d_ws, size_t ws_size,
                              hipStream_t stream) {
  const float* in  = (const float*)d_in[0];
  float*       out = (float*)d_out;
  const int nmat = in_sizes[0] / (kN * kN);   // 4096
  sinkhorn128_kernel<<<dim3(nmat), dim3(kN), 0, stream>>>(in, out);
}